// MoLELayer_39273180954889
// MI455X (gfx1250) — compile-verified
//
#include <hip/hip_runtime.h>

typedef __attribute__((ext_vector_type(2))) float v2f;
typedef __attribute__((ext_vector_type(8))) float v8f;
typedef __attribute__((ext_vector_type(4))) unsigned int v4u;
typedef __attribute__((ext_vector_type(8))) int v8i;
typedef __attribute__((ext_vector_type(4))) int v4i;

// Problem dims (match reference)
constexpr int BB   = 4;
constexpr int S    = 4096;
constexpr int D    = 2048;   // K
constexpr int OUTD = 2048;   // N
constexpr int E    = 8;
constexpr int R    = 16;
constexpr int M    = BB * S; // 16384 tokens
constexpr float SCALING = 32.0f / 16.0f; // alpha / r = 2.0

// GEMM tiling
constexpr int KB    = 32;          // K chunk staged in LDS
constexpr int LDST  = 36;          // padded LDS row stride (floats)
constexpr int TILEF = 128 * LDST;  // floats per 128x32 tile in LDS (4608)
constexpr int TILEB = TILEF * 4;   // 18432 bytes
constexpr int BUFF  = 2 * TILEF;   // floats per buffer (X tile + W tile)

#if __has_builtin(__builtin_amdgcn_tensor_load_to_lds) && __has_builtin(__builtin_amdgcn_s_wait_tensorcnt)
#define USE_TDM 1
#else
#define USE_TDM 0
#endif

#if USE_TDM
// LDS byte offset of a __shared__ object: addrspacecast generic->AS(3), ptrtoint.
__device__ __forceinline__ unsigned lds_addr_u32(const void* p) {
    return (unsigned)(unsigned long long)(const __attribute__((address_space(3))) char*)p;
}

// TDM: one 2D tile load, 128 rows x 32 f32, memory row stride D floats,
// deposited in LDS with 4 pad dwords after every 32 dwords (-> stride 36).
__device__ __forceinline__ void tdm_load_tile(const float* gsrc, unsigned ldsByte) {
    const unsigned long long ga = (unsigned long long)gsrc;
    v4u g0;
    g0.x = 1u;                                  // count=1 (valid user descriptor)
    g0.y = ldsByte;                             // lds_addr (bytes)
    g0.z = (unsigned)ga;                        // global_addr[31:0]
    g0.w = (unsigned)((ga >> 32) & 0x01FFFFFFu) // global_addr[56:32]
         | (2u << 30);                          // type = 2 ("image")
    v8i g1;
    g1[0] = (int)((2u << 16)     // data_size = 4 bytes
                | (1u << 20)     // pad_enable
                | (4u << 22)     // pad_interval: 32 dwords
                | (3u << 25));   // pad_amount:   4 dwords
    g1[1] = (int)((unsigned)KB << 16);   // tensor_dim0 = 32 (low16 -> bits 63:48)
    g1[2] = (int)(128u << 16);           // tensor_dim0 hi=0 | tensor_dim1 = 128
    g1[3] = (int)((unsigned)KB << 16);   // tensor_dim1 hi=0 | tile_dim0 = 32
    g1[4] = 128;                         // tile_dim1 = 128 | tile_dim2 = 0
    g1[5] = D;                           // tensor_dim0_stride = 2048
    g1[6] = 0;
    g1[7] = 0;
    const v4i z4 = {0, 0, 0, 0};
    const v8i z8 = {0, 0, 0, 0, 0, 0, 0, 0};
    __builtin_amdgcn_tensor_load_to_lds(g0, g1, z4, z4, z8, 0);
}
#endif

// ---------------------------------------------------------------------------
// Kernel 1: base_out = X @ W^T + b   via V_WMMA_F32_16X16X4_F32
//   WG: 256 thr = 8 waves, tile 128x128; wave grid 4(M) x 2(N);
//   each wave: 2x4 tiles of 16x16 -> 8 v8f accumulators.
//   K staged in LDS by the Tensor Data Mover, double buffered.
// ---------------------------------------------------------------------------
__global__ __launch_bounds__(256) void base_gemm_wmma(
    const float* __restrict__ X,      // [M, D]
    const float* __restrict__ W,      // [OUTD, D]
    const float* __restrict__ bias,   // [OUTD]
    float* __restrict__ out)          // [M, OUTD]
{
#if USE_TDM
    __shared__ float smem[2 * BUFF];  // 73728 B: [buf0 X][buf0 W][buf1 X][buf1 W]
#else
    __shared__ float smem[BUFF];      // 36864 B: [X][W]
#endif

    const int tid   = threadIdx.x;
    const int lane  = tid & 31;
    const int wave  = tid >> 5;
    const int waveM = wave >> 1;          // 0..3
    const int waveN = wave & 1;           // 0..1
    const int hlf   = lane >> 4;          // 0/1
    const int row   = lane & 15;

    const int mBlock = blockIdx.y * 128;
    const int nBlock = blockIdx.x * 128;

    v8f acc[2][4] = {};

#if USE_TDM
    const unsigned ldsBase = lds_addr_u32(smem);
    if (wave == 0) tdm_load_tile(&X[(size_t)mBlock * D], ldsBase);
    if (wave == 1) tdm_load_tile(&W[(size_t)nBlock * D], ldsBase + TILEB);
#endif

    for (int kb = 0; kb < D; kb += KB) {
#if USE_TDM
        const int  cur  = (kb / KB) & 1;
        const bool more = (kb + KB) < D;
        if (more) {
            const unsigned nb = ldsBase + (unsigned)((cur ^ 1) * (BUFF * 4));
            if (wave == 0) tdm_load_tile(&X[(size_t)mBlock * D + kb + KB], nb);
            if (wave == 1) tdm_load_tile(&W[(size_t)nBlock * D + kb + KB], nb + TILEB);
        }
        if (wave < 2) {
            if (more) __builtin_amdgcn_s_wait_tensorcnt(1);  // current pair done, next in flight
            else      __builtin_amdgcn_s_wait_tensorcnt(0);  // drain
        }
        __syncthreads();   // current buffer valid for all waves
        const float* sX = &smem[cur * BUFF];
        const float* sW = sX + TILEF;
#else
        // Fallback: stage through VGPRs.
        #pragma unroll
        for (int j = 0; j < 4; ++j) {
            const int flat = tid + 256 * j;      // 0..1023
            const int r    = flat >> 3;          // 0..127
            const int c    = (flat & 7) << 2;    // 0,4,...,28
            const float4 vx = *reinterpret_cast<const float4*>(
                &X[(size_t)(mBlock + r) * D + kb + c]);
            const float4 vw = *reinterpret_cast<const float4*>(
                &W[(size_t)(nBlock + r) * D + kb + c]);
            *reinterpret_cast<float4*>(&smem[r * LDST + c])         = vx;
            *reinterpret_cast<float4*>(&smem[TILEF + r * LDST + c]) = vw;
        }
        __syncthreads();
        const float* sX = smem;
        const float* sW = smem + TILEF;
#endif

        // --- 8 K-steps of 4 on the matrix pipe ---
        #pragma unroll
        for (int ks = 0; ks < KB / 4; ++ks) {
            const int cb = ks * 4 + 2 * hlf;   // per-lane K offset (even -> b64 aligned)
            v2f afr[2], bfr[4];
            #pragma unroll
            for (int mi = 0; mi < 2; ++mi)
                afr[mi] = *reinterpret_cast<const v2f*>(
                    &sX[(waveM * 32 + mi * 16 + row) * LDST + cb]);
            #pragma unroll
            for (int ni = 0; ni < 4; ++ni)
                bfr[ni] = *reinterpret_cast<const v2f*>(
                    &sW[(waveN * 64 + ni * 16 + row) * LDST + cb]);
            #pragma unroll
            for (int mi = 0; mi < 2; ++mi)
                #pragma unroll
                for (int ni = 0; ni < 4; ++ni)
                    acc[mi][ni] = __builtin_amdgcn_wmma_f32_16x16x4_f32(
                        false, afr[mi], false, bfr[ni],
                        (short)0, acc[mi][ni], false, false);
        }
        __syncthreads();   // all waves done reading this buffer
    }

    // --- epilogue: add bias, store (C layout: VGPR v -> row v + 8*hlf, col = lane&15) ---
    #pragma unroll
    for (int mi = 0; mi < 2; ++mi) {
        #pragma unroll
        for (int ni = 0; ni < 4; ++ni) {
            const int n  = nBlock + waveN * 64 + ni * 16 + row;
            const float bb = bias[n];
            #pragma unroll
            for (int v = 0; v < 8; ++v) {
                const int m = mBlock + waveM * 32 + mi * 16 + v + 8 * hlf;
                out[(size_t)m * OUTD + n] = acc[mi][ni][v] + bb;
            }
        }
    }
}

// ---------------------------------------------------------------------------
// Kernel 2: fused router (argmax of logits == argmax of softmax) + LoRA
// ---------------------------------------------------------------------------
__global__ __launch_bounds__(256) void router_lora(
    const float* __restrict__ x,       // [M, D]
    const float* __restrict__ routerW, // [E, D]
    const float* __restrict__ A,       // [E, R, D]
    const float* __restrict__ Bw,      // [E, OUTD, R]
    float* __restrict__ out)           // [M, OUTD] (accumulate)
{
    __shared__ float sLogit[E];
    __shared__ float sH[R];
    __shared__ int   sSel;

    const int t    = blockIdx.x;
    const int tid  = threadIdx.x;
    const int lane = tid & 31;
    const int wave = tid >> 5;   // 0..7 == expert id / (r-pair) id
    const float* xt = x + (size_t)t * D;

    // logits: wave e reduces x . routerW[e] over D
    {
        const float* wr = routerW + (size_t)wave * D;
        float acc = 0.f;
        for (int d = lane; d < D; d += 32) acc += xt[d] * wr[d];
        #pragma unroll
        for (int m = 16; m; m >>= 1) acc += __shfl_xor(acc, m, 32);
        if (lane == 0) sLogit[wave] = acc;
    }
    __syncthreads();
    if (tid == 0) {
        int best = 0; float bv = sLogit[0];
        #pragma unroll
        for (int e = 1; e < E; ++e) {
            const float v = sLogit[e];
            if (v > bv) { bv = v; best = e; }   // first-max tie-break, like argmax
        }
        sSel = best;
    }
    __syncthreads();
    const int e = sSel;

    // h[r] = x . A[e][r]  (wave w does r = 2w, 2w+1)
    {
        const float* a0 = A + ((size_t)e * R + 2 * wave) * D;
        const float* a1 = a0 + D;
        float h0 = 0.f, h1 = 0.f;
        for (int d = lane; d < D; d += 32) {
            const float xv = xt[d];
            h0 += xv * a0[d];
            h1 += xv * a1[d];
        }
        #pragma unroll
        for (int m = 16; m; m >>= 1) {
            h0 += __shfl_xor(h0, m, 32);
            h1 += __shfl_xor(h1, m, 32);
        }
        if (lane == 0) { sH[2 * wave] = h0; sH[2 * wave + 1] = h1; }
    }
    __syncthreads();

    float h[R];
    #pragma unroll
    for (int r = 0; r < R; ++r) h[r] = sH[r];

    // up-projection: out[t][o] += SCALING * (Bw[e][o][:] . h)
    float* ot = out + (size_t)t * OUTD;
    #pragma unroll
    for (int j = 0; j < OUTD / 256; ++j) {
        const int o = tid + 256 * j;
        const float* bwo = Bw + ((size_t)e * OUTD + o) * R;
        float acc = 0.f;
        #pragma unroll
        for (int r = 0; r < R; ++r) acc += h[r] * bwo[r];
        ot[o] += SCALING * acc;
    }
}

// ---------------------------------------------------------------------------
extern "C" void kernel_launch(void* const* d_in, const int* in_sizes, int n_in,
                              void* d_out, int out_size, void* d_ws, size_t ws_size,
                              hipStream_t stream) {
    (void)in_sizes; (void)n_in; (void)out_size; (void)d_ws; (void)ws_size;
    const float* x    = (const float*)d_in[0];
    const float* Wb   = (const float*)d_in[1];
    const float* bb   = (const float*)d_in[2];
    const float* rW   = (const float*)d_in[3];
    const float* A    = (const float*)d_in[4];
    const float* Bw   = (const float*)d_in[5];
    float* out        = (float*)d_out;

    dim3 grid(OUTD / 128, M / 128);   // 16 x 128 workgroups
    base_gemm_wmma<<<grid, 256, 0, stream>>>(x, Wb, bb, out);
    router_lora<<<M, 256, 0, stream>>>(x, rW, A, Bw, out);
}